// KeypPointBboxNet_52055003628264
// MI455X (gfx1250) — compile-verified
//
#include <hip/hip_runtime.h>
#include <cstdint>
#include <cstddef>

// Problem dims fixed by the reference harness.
#define BDIM 8
#define CDIM 256
#define HDIM 128
#define WDIM 128
#define SPIX (HDIM * WDIM)   // 16384 pixels per (b, c) plane

typedef float v4f __attribute__((ext_vector_type(4)));

// ---------------------------------------------------------------------------
// Pass 1: BCHW -> BHWC transpose (per batch: 256 x 16384 -> 16384 x 256).
// 32x32 tile per 256-thread block.
//   load : gfx1250 GLOBAL_LOAD_ASYNC_TO_LDS_B128 — each lane DMAs 16 B of
//          src[c0 + t/8][s0 + (t%8)*4 ..+3] straight into LDS (no VGPR
//          staging), tracked by ASYNCcnt, drained with s_wait_asynccnt 0.
//   store: thread t writes dst[s0 + t/8][c0 + (t%8)*4 ..+3] as one b128.
// LDS rows padded to 36 floats (144 B) so every async B128 LDS target is
// 16 B aligned. Transposed read is 2-way bank conflicted — irrelevant vs HBM.
// Source is read exactly once -> th:TH_LOAD_NT; dst is re-read by the gather
// pass, so regular (RT) stores keep it hot in the 192 MB L2.
// ---------------------------------------------------------------------------
__global__ __launch_bounds__(256) void transpose_bchw_bhwc(
    const float* __restrict__ src, float* __restrict__ dst) {
  __shared__ float tile[32][36];
  const int b  = blockIdx.z;
  const int s0 = blockIdx.x * 32;   // pixel tile origin
  const int c0 = blockIdx.y * 32;   // channel tile origin
  const float* sb = src + (size_t)b * CDIM * SPIX;
  float*       db = dst + (size_t)b * SPIX * CDIM;

  const int t    = (int)threadIdx.x;  // 0..255
  const int row  = t >> 3;            // 0..31
  const int quad = (t & 7) * 4;       // 0,4,...,28

  // Per-lane source byte offset within this batch's plane (< 16 MiB, fits i32)
  const unsigned src_off = (unsigned)((c0 + row) * SPIX + s0 + quad) * 4u;
  // Generic shared pointers carry the LDS offset in addr[31:0].
  const unsigned lds_addr = (unsigned)(uintptr_t)&tile[row][quad];

  // Async memory -> LDS copy, 16 B per lane (gfx1250 VGLOBAL async encoding,
  // GVS mode: SGPR64 base + per-lane 32-bit byte offset).
  asm volatile("global_load_async_to_lds_b128 %0, %1, %2 offset:0 th:TH_LOAD_NT"
               :
               : "v"(lds_addr), "v"(src_off), "s"(sb)
               : "memory");
  asm volatile("s_wait_asynccnt 0" ::: "memory");
  __syncthreads();

  v4f o;
  o.x = tile[quad + 0][row];
  o.y = tile[quad + 1][row];
  o.z = tile[quad + 2][row];
  o.w = tile[quad + 3][row];
  *(v4f*)&db[(size_t)(s0 + row) * CDIM + c0 + quad] = o;
}

// ---------------------------------------------------------------------------
// Shared coordinate / weight math, replicating the reference exactly.
// ---------------------------------------------------------------------------
struct SampleGeom {
  int b, hl, wl, hh, wh;
  float w1, w2, w3, w4;
};

__device__ __forceinline__ SampleGeom compute_geom(
    const float* __restrict__ rois, const float* __restrict__ offs,
    int n, int p, int num_point, float inv_stride) {
  const float r0 = rois[n * 5 + 0];
  const float x1 = rois[n * 5 + 1];
  const float y1 = rois[n * 5 + 2];
  const float x2 = rois[n * 5 + 3];
  const float y2 = rois[n * 5 + 4];
  const float ox = offs[n * (2 * num_point) + 2 * p + 0];
  const float oy = offs[n * (2 * num_point) + 2 * p + 1];

  const float cx = (x1 + x2) * 0.5f;
  const float cy = (y1 + y2) * 0.5f;
  const float w_ = x2 - x1 + 1.0f;
  const float h_ = y2 - y1 + 1.0f;
  float wf = (cx + (ox * w_) * 0.1f) * inv_stride;   // x (w coord)
  float hf = (cy + (oy * h_) * 0.1f) * inv_stride;   // y (h coord)

  // Reference edge handling: clip floor to [0, dim-1]; only when the clipped
  // low index is at dim-1 collapse the high index AND the coordinate.
  float hlf = floorf(hf);
  hlf = fminf(fmaxf(hlf, 0.0f), (float)(HDIM - 1));
  float wlf = floorf(wf);
  wlf = fminf(fmaxf(wlf, 0.0f), (float)(WDIM - 1));

  SampleGeom g;
  g.hl = (int)hlf;
  g.wl = (int)wlf;
  if (hlf >= (float)(HDIM - 1)) { g.hh = g.hl; hf = hlf; } else { g.hh = g.hl + 1; }
  if (wlf >= (float)(WDIM - 1)) { g.wh = g.wl; wf = wlf; } else { g.wh = g.wl + 1; }

  const float lh = hf - hlf;          // may be negative (extrapolation) — matches ref
  const float lw = wf - wlf;
  const float ch = 1.0f - lh;
  const float cw = 1.0f - lw;
  g.w1 = ch * cw;
  g.w2 = ch * lw;
  g.w3 = lh * cw;
  g.w4 = lh * lw;
  g.b  = (int)r0;
  return g;
}

// ---------------------------------------------------------------------------
// Pass 2 (fast path): gather from BHWC copy. One wave32 per sample point.
// Each corner is a 1 KB contiguous channel vector: lane reads a b128 chunk at
// c = lane*4 and c = 128 + lane*4, so every wave load is fully packed
// cachelines. Output stored nontemporally (never re-read on device).
// ---------------------------------------------------------------------------
__global__ __launch_bounds__(256) void deform_gather_bhwc(
    const float* __restrict__ fmt, const float* __restrict__ rois,
    const float* __restrict__ offs, const int* __restrict__ stride_p,
    float* __restrict__ out, int n_rois, int num_point) {
  const int gthread = blockIdx.x * 256 + (int)threadIdx.x;
  const int wave = gthread >> 5;
  const int lane = threadIdx.x & 31;
  const int total = n_rois * num_point;
  if (wave >= total) return;
  const int n = wave / num_point;
  const int p = wave - n * num_point;

  const float inv_stride = 1.0f / (float)(*stride_p);
  const SampleGeom g = compute_geom(rois, offs, n, p, num_point, inv_stride);

  const float* base = fmt + (size_t)g.b * SPIX * CDIM;
  const float* p1 = base + ((size_t)g.hl * WDIM + g.wl) * CDIM;
  const float* p2 = base + ((size_t)g.hl * WDIM + g.wh) * CDIM;
  const float* p3 = base + ((size_t)g.hh * WDIM + g.wl) * CDIM;
  const float* p4 = base + ((size_t)g.hh * WDIM + g.wh) * CDIM;
  float* o = out + (size_t)wave * CDIM;

#pragma unroll
  for (int k = 0; k < 2; ++k) {
    const int c = k * 128 + lane * 4;
    const v4f a  = *(const v4f*)(p1 + c);
    const v4f b  = *(const v4f*)(p2 + c);
    const v4f cc = *(const v4f*)(p3 + c);
    const v4f d  = *(const v4f*)(p4 + c);
    v4f r = g.w1 * a + g.w2 * b + g.w3 * cc + g.w4 * d;
    __builtin_nontemporal_store(r, (v4f*)(o + c));
  }
}

// ---------------------------------------------------------------------------
// Fallback: direct gather from BCHW (only if workspace too small for the
// transposed copy). Scattered 4B loads, correct but slower.
// ---------------------------------------------------------------------------
__global__ __launch_bounds__(256) void deform_gather_bchw(
    const float* __restrict__ fm, const float* __restrict__ rois,
    const float* __restrict__ offs, const int* __restrict__ stride_p,
    float* __restrict__ out, int n_rois, int num_point) {
  const int gthread = blockIdx.x * 256 + (int)threadIdx.x;
  const int wave = gthread >> 5;
  const int lane = threadIdx.x & 31;
  const int total = n_rois * num_point;
  if (wave >= total) return;
  const int n = wave / num_point;
  const int p = wave - n * num_point;

  const float inv_stride = 1.0f / (float)(*stride_p);
  const SampleGeom g = compute_geom(rois, offs, n, p, num_point, inv_stride);

  const float* base = fm + (size_t)g.b * CDIM * SPIX;
  const size_t i1 = (size_t)g.hl * WDIM + g.wl;
  const size_t i2 = (size_t)g.hl * WDIM + g.wh;
  const size_t i3 = (size_t)g.hh * WDIM + g.wl;
  const size_t i4 = (size_t)g.hh * WDIM + g.wh;
  float* o = out + (size_t)wave * CDIM;

#pragma unroll
  for (int k = 0; k < 8; ++k) {
    const int c = lane + k * 32;
    const float* cp = base + (size_t)c * SPIX;
    const float v = g.w1 * cp[i1] + g.w2 * cp[i2] + g.w3 * cp[i3] + g.w4 * cp[i4];
    __builtin_nontemporal_store(v, &o[c]);
  }
}

// ---------------------------------------------------------------------------
// Host-side launcher. Inputs (setup_inputs order):
//   d_in[0] feat_map f32 [8,256,128,128]
//   d_in[1] rois     f32 [N,5]
//   d_in[2] offset   f32 [N,2P]
//   d_in[3] stride   i32 scalar (read on device)
//   d_in[4] num_point i32 scalar (value derived on host from sizes)
// Output: f32 [N,P,256]
// ---------------------------------------------------------------------------
extern "C" void kernel_launch(void* const* d_in, const int* in_sizes, int n_in,
                              void* d_out, int out_size, void* d_ws, size_t ws_size,
                              hipStream_t stream) {
  const float* feat   = (const float*)d_in[0];
  const float* rois   = (const float*)d_in[1];
  const float* offs   = (const float*)d_in[2];
  const int* stride_p = (const int*)d_in[3];
  float* out = (float*)d_out;

  const int n_rois    = in_sizes[1] / 5;
  const int total     = out_size / CDIM;       // N * P sample points
  const int num_point = total / n_rois;

  const int gather_threads = total * 32;       // one wave32 per sample
  const int gather_blocks  = (gather_threads + 255) / 256;

  const size_t need = (size_t)BDIM * SPIX * CDIM * sizeof(float);  // 128 MiB
  if (ws_size >= need) {
    float* fmt = (float*)d_ws;                  // BHWC copy
    dim3 tb(256, 1, 1);
    dim3 tg(SPIX / 32, CDIM / 32, BDIM);
    transpose_bchw_bhwc<<<tg, tb, 0, stream>>>(feat, fmt);
    deform_gather_bhwc<<<gather_blocks, 256, 0, stream>>>(
        fmt, rois, offs, stride_p, out, n_rois, num_point);
  } else {
    deform_gather_bchw<<<gather_blocks, 256, 0, stream>>>(
        feat, rois, offs, stride_p, out, n_rois, num_point);
  }
}